// BertForSequenceClassification_three_26800595927388
// MI455X (gfx1250) — compile-verified
//
#include <hip/hip_runtime.h>
#include <math.h>

typedef __attribute__((ext_vector_type(2))) float v2f;
typedef __attribute__((ext_vector_type(4))) float f4;
typedef __attribute__((ext_vector_type(8))) float v8f;

#define BB 32
#define SS 512
#define HH 768
#define FDD 64
#define PP 30
#define PPAD 32

// workspace layout in floats
#define WS_SCALE 0
#define WS_PART  64                         // 2048 block partials
#define WS_TEXT  4096                       // B*S*PPAD = 524288
#define WS_AUDIO (4096 + BB*SS*PPAD)        // B*S*PPAD
#define WS_ROW0  (4096 + 2*BB*SS*PPAD)      // B*S fusion_raw at s==0

// ---------------- K1: projections + sum-of-squares partials ----------------
__global__ __launch_bounds__(256) void k_proj(const float* __restrict__ hs,
                                              const float* __restrict__ ad,
                                              const float* __restrict__ Wt,
                                              const float* __restrict__ Wa,
                                              float* __restrict__ ws) {
  __shared__ float wred[8];
  int wave = threadIdx.x >> 5;
  int lane = threadIdx.x & 31;
  int row  = blockIdx.x * 8 + wave;           // 0..B*S-1
  float* tex = ws + WS_TEXT;
  float* aud = ws + WS_AUDIO;
  float acc = 0.f, acca = 0.f;
  if (lane < PP) {
    const float* hrow = hs + (size_t)row * HH;
    const float* wrow = Wt + (size_t)lane * HH;
    for (int k = 0; k < HH; k += 4) {
      f4 hv = *(const f4*)(hrow + k);
      f4 wv = *(const f4*)(wrow + k);
      acc += hv.x * wv.x + hv.y * wv.y + hv.z * wv.z + hv.w * wv.w;
    }
    const float* arow  = ad + (size_t)row * FDD;
    const float* warow = Wa + (size_t)lane * FDD;
    for (int k = 0; k < FDD; k += 4) {
      f4 av = *(const f4*)(arow + k);
      f4 wv = *(const f4*)(warow + k);
      acca += av.x * wv.x + av.y * wv.y + av.z * wv.z + av.w * wv.w;
    }
  }
  tex[(size_t)row * PPAD + lane] = acc;       // lanes 30,31 write zero padding
  aud[(size_t)row * PPAD + lane] = acca;
  float ssq = acc * acc;
  for (int m = 16; m >= 1; m >>= 1) ssq += __shfl_xor(ssq, m, 32);
  if (lane == 0) wred[wave] = ssq;
  __syncthreads();
  if (threadIdx.x == 0) {
    float t = 0.f;
    for (int w = 0; w < 8; ++w) t += wred[w];
    ws[WS_PART + blockIdx.x] = t;
  }
}

// ---------------- K1b: reduce partials -> scale = 1/sqrt(sumsq) ----------------
__global__ __launch_bounds__(256) void k_norm(float* __restrict__ ws) {
  __shared__ float red[256];
  float s = 0.f;
  for (int i = threadIdx.x; i < 2048; i += 256) s += ws[WS_PART + i];
  red[threadIdx.x] = s;
  __syncthreads();
  for (int st = 128; st > 0; st >>= 1) {
    if (threadIdx.x < st) red[threadIdx.x] += red[threadIdx.x + st];
    __syncthreads();
  }
  if (threadIdx.x == 0) ws[WS_SCALE] = 1.0f / sqrtf(red[0]);
}

// ---------------- K2: WMMA Gram tiles -> text_att1 & fusion_att1 ----------------
__global__ __launch_bounds__(256) void k_att(float* __restrict__ ws,
                                             const float* __restrict__ p_tw,
                                             const float* __restrict__ p_aw,
                                             const float* __restrict__ p_fb,
                                             float* __restrict__ out) {
  const float* tex = ws + WS_TEXT;
  const float* aud = ws + WS_AUDIO;
  float* row0 = ws + WS_ROW0;
  float scale = ws[WS_SCALE];
  float tw = p_tw[0], aw = p_aw[0], fb = p_fb[0];

  int lane = threadIdx.x & 31;
  int lm = lane & 15;
  int hi = lane >> 4;
  int tile = blockIdx.x * 8 + (threadIdx.x >> 5);   // 0..32767
  int b  = tile >> 10;
  int ti = (tile >> 5) & 31;
  int tj = tile & 31;

  // f32 16x16x4 A-fragment: lanes 0-15 -> M=lane, K pair {0,1}; lanes 16-31 -> K pair {2,3}
  const float* Ai = tex + ((size_t)(b * SS) + ti * 16 + lm) * PPAD + 2 * hi;
  const float* Bj = tex + ((size_t)(b * SS) + tj * 16 + lm) * PPAD + 2 * hi;
  v8f ct = {};
  for (int ks = 0; ks < 8; ++ks) {
    v2f a  = *(const v2f*)(Ai + ks * 4);
    v2f bb = *(const v2f*)(Bj + ks * 4);
    ct = __builtin_amdgcn_wmma_f32_16x16x4_f32(false, a, false, bb, (short)0, ct,
                                               false, false);
  }
  const float* Ca = aud + ((size_t)(b * SS) + ti * 16 + lm) * PPAD + 2 * hi;
  const float* Da = aud + ((size_t)(b * SS) + tj * 16 + lm) * PPAD + 2 * hi;
  v8f ca = {};
  for (int ks = 0; ks < 8; ++ks) {
    v2f a  = *(const v2f*)(Ca + ks * 4);
    v2f bb = *(const v2f*)(Da + ks * 4);
    ca = __builtin_amdgcn_wmma_f32_16x16x4_f32(false, a, false, bb, (short)0, ca,
                                               false, false);
  }

  float* ot = out + (size_t)BB * HH;                          // text_att1 base
  float* of = ot + (size_t)BB * SS * SS;                      // fusion_att1 base
  size_t obase = (size_t)b * SS * SS;
  for (int r = 0; r < 8; ++r) {
    int rowi = ti * 16 + r + 8 * hi;                          // C/D layout row
    int colj = tj * 16 + lm;
    float tval = ct[r] * scale;                               // text.text / norm
    float ta = tval > 0.f ? tval : 0.f;                       // text_att1
    float av = ca[r];
    float aa = av > 0.f ? av : 0.f;                           // audio_att
    float fraw = tw * ta + aw * aa + fb;                      // fusion_att_raw
    float f1 = fraw > 0.f ? fraw : 0.f;                       // fusion_att1
    size_t idx = obase + (size_t)rowi * SS + colj;
    ot[idx] = ta;
    of[idx] = f1;
    if (ti == 0 && r == 0 && hi == 0) row0[b * SS + colj] = fraw;  // s==0 logits
  }
}

// ---------------- K3: softmax row0 + context + dense + LayerNorm ----------------
__global__ __launch_bounds__(256) void k_head(const float* __restrict__ hs,
                                              const float* __restrict__ am,
                                              const float* __restrict__ Wd,
                                              const float* __restrict__ bd,
                                              const float* __restrict__ lw,
                                              const float* __restrict__ lb,
                                              const float* __restrict__ ws,
                                              float* __restrict__ out) {
  __shared__ float att[SS];
  __shared__ float fr[HH];
  __shared__ float red[256];
  int b = blockIdx.x;
  int tid = threadIdx.x;
  const float* row0 = ws + WS_ROW0 + (size_t)b * SS;
  const float* amb  = am + (size_t)b * SS;                    // mask[b,0,0,:]
  // mask_t adds mask[b,0] uniformly to row 0 -> softmax-invariant, omitted.
  float l0 = row0[tid] + amb[tid];
  float l1 = row0[tid + 256] + amb[tid + 256];
  red[tid] = fmaxf(l0, l1);
  __syncthreads();
  for (int st = 128; st > 0; st >>= 1) {
    if (tid < st) red[tid] = fmaxf(red[tid], red[tid + st]);
    __syncthreads();
  }
  float m = red[0];
  __syncthreads();
  float e0 = expf(l0 - m), e1 = expf(l1 - m);
  red[tid] = e0 + e1;
  __syncthreads();
  for (int st = 128; st > 0; st >>= 1) {
    if (tid < st) red[tid] += red[tid + st];
    __syncthreads();
  }
  float inv = 1.0f / red[0];
  __syncthreads();
  att[tid] = e0 * inv;
  att[tid + 256] = e1 * inv;
  __syncthreads();

  // fusion_data[b,0,:] = att . hidden[b] + hidden[b,0,:]
  const float* hb = hs + (size_t)b * SS * HH;
  int h0 = tid, h1 = tid + 256, h2 = tid + 512;
  float a0 = 0.f, a1 = 0.f, a2 = 0.f;
  for (int t = 0; t < SS; ++t) {
    float a = att[t];
    const float* hr = hb + (size_t)t * HH;
    a0 += a * hr[h0];
    a1 += a * hr[h1];
    a2 += a * hr[h2];
  }
  fr[h0] = a0 + hb[h0];
  fr[h1] = a1 + hb[h1];
  fr[h2] = a2 + hb[h2];
  __syncthreads();

  // h = fusion_row @ W_dense^T + b_dense
  float o[3];
  for (int j = 0; j < 3; ++j) {
    int oi = tid + j * 256;
    const float* wrow = Wd + (size_t)oi * HH;
    float acc = 0.f;
    for (int k = 0; k < HH; k += 4) {
      f4 wv = *(const f4*)(wrow + k);
      acc += wv.x * fr[k] + wv.y * fr[k + 1] + wv.z * fr[k + 2] + wv.w * fr[k + 3];
    }
    o[j] = acc + bd[oi];
  }
  __syncthreads();
  // LayerNorm (two-pass)
  red[tid] = o[0] + o[1] + o[2];
  __syncthreads();
  for (int st = 128; st > 0; st >>= 1) {
    if (tid < st) red[tid] += red[tid + st];
    __syncthreads();
  }
  float u = red[0] * (1.0f / HH);
  __syncthreads();
  float d0 = o[0] - u, d1 = o[1] - u, d2 = o[2] - u;
  red[tid] = d0 * d0 + d1 * d1 + d2 * d2;
  __syncthreads();
  for (int st = 128; st > 0; st >>= 1) {
    if (tid < st) red[tid] += red[tid + st];
    __syncthreads();
  }
  float rstd = 1.0f / sqrtf(red[0] * (1.0f / HH) + 1e-12f);
  out[(size_t)b * HH + h0] = lw[h0] * d0 * rstd + lb[h0];
  out[(size_t)b * HH + h1] = lw[h1] * d1 * rstd + lb[h1];
  out[(size_t)b * HH + h2] = lw[h2] * d2 * rstd + lb[h2];
}

extern "C" void kernel_launch(void* const* d_in, const int* in_sizes, int n_in,
                              void* d_out, int out_size, void* d_ws, size_t ws_size,
                              hipStream_t stream) {
  const float* hs = (const float*)d_in[0];   // hidden_states (B,S,H)
  const float* ad = (const float*)d_in[1];   // audio_data (B,S,FD)
  const float* am = (const float*)d_in[2];   // attention_mask (B,1,1,S)
  const float* Wt = (const float*)d_in[3];   // (P,H)
  const float* Wa = (const float*)d_in[4];   // (P,FD)
  const float* tw = (const float*)d_in[5];
  const float* aw = (const float*)d_in[6];
  const float* fb = (const float*)d_in[7];
  const float* Wd = (const float*)d_in[8];   // (H,H)
  const float* bd = (const float*)d_in[9];
  const float* lw = (const float*)d_in[10];
  const float* lb = (const float*)d_in[11];
  float* out = (float*)d_out;
  float* ws  = (float*)d_ws;

  k_proj<<<2048, 256, 0, stream>>>(hs, ad, Wt, Wa, ws);
  k_norm<<<1, 256, 0, stream>>>(ws);
  k_att<<<4096, 256, 0, stream>>>(ws, tw, aw, fb, out);
  k_head<<<32, 256, 0, stream>>>(hs, am, Wd, bd, lw, lb, ws, out);
}